// PhaseModulatedConvBlock_5428838662527
// MI455X (gfx1250) — compile-verified
//
#include <hip/hip_runtime.h>
#include <math.h>

typedef __attribute__((ext_vector_type(2))) float v2f;
typedef __attribute__((ext_vector_type(8))) float v8f;

#define HID 128
#define NH  4

__device__ __forceinline__ float wred_add(float v) {
#pragma unroll
  for (int off = 16; off > 0; off >>= 1) v += __shfl_xor(v, off, 32);
  return v;
}

// ---------------------------------------------------------------- fill
__global__ void fill_kernel(float* __restrict__ p, size_t n, float v) {
  size_t i = (size_t)blockIdx.x * blockDim.x + threadIdx.x;
  size_t stride = (size_t)gridDim.x * blockDim.x;
  for (; i < n; i += stride) p[i] = v;
}

// ---------------------------------------------------- GEMM: Y = X(N,128) @ W(128,128) + bias
// One wave per 16-row strip; 8 column tiles of 16; f32 WMMA 16x16x4, K swept in steps of 4.
__global__ void gemm128_wmma(const float* __restrict__ X, const float* __restrict__ W,
                             const float* __restrict__ bias, float* __restrict__ Y,
                             int n_rows) {
  const int wave = threadIdx.x >> 5;
  const int lane = threadIdx.x & 31;
  const int row0 = (blockIdx.x * 8 + wave) * 16;
  if (row0 >= n_rows) return;
  const int lm = lane & 15;
  const int hi = lane >> 4;  // half-wave selects K pair

  int arow = row0 + lm;
  if (arow >= n_rows) arow = n_rows - 1;  // clamp loads; stores guarded below
  const float* xrow = X + (size_t)arow * HID + 2 * hi;

  v8f c[8];
#pragma unroll
  for (int t = 0; t < 8; ++t) {
    float bv = bias ? bias[t * 16 + lm] : 0.0f;  // bias depends on column only
    c[t] = (v8f){bv, bv, bv, bv, bv, bv, bv, bv};
  }

  for (int k = 0; k < HID; k += 4) {
    v2f a;
    a.x = xrow[k];
    a.y = xrow[k + 1];
    const float* wk = W + (size_t)(k + 2 * hi) * HID + lm;
#pragma unroll
    for (int t = 0; t < 8; ++t) {
      v2f b;
      b.x = wk[t * 16];
      b.y = wk[t * 16 + HID];
      c[t] = __builtin_amdgcn_wmma_f32_16x16x4_f32(false, a, false, b, (short)0, c[t],
                                                   false, false);
    }
  }

#pragma unroll
  for (int r = 0; r < 8; ++r) {
    int rr = row0 + r + 8 * hi;  // C/D layout: vgpr r -> M = r (+8 for upper half-wave)
    if (rr < n_rows) {
      float* y = Y + (size_t)rr * HID + lm;
#pragma unroll
      for (int t = 0; t < 8; ++t) y[t * 16] = c[t][r];
    }
  }
}

// ---------------------------------------------------- GATv2 edge pass 1: per-(dst,head) max logit
__global__ void gat_max_kernel(const int* __restrict__ src, const int* __restrict__ dst, int E,
                               const float* __restrict__ XL, const float* __restrict__ XR,
                               const float* __restrict__ att, float* __restrict__ MAXB) {
  int e = (int)(((size_t)blockIdx.x * blockDim.x + threadIdx.x) >> 5);
  int lane = threadIdx.x & 31;
  if (e >= E) return;
  int s = src[e], d = dst[e];
  const float* xl = XL + (size_t)s * HID;
  const float* xr = XR + (size_t)d * HID;
  float lg0, lg1, lg2, lg3;
  {
    float z;
    z = xl[lane] + xr[lane];             z = z > 0.f ? z : 0.2f * z; lg0 = wred_add(z * att[lane]);
    z = xl[32 + lane] + xr[32 + lane];   z = z > 0.f ? z : 0.2f * z; lg1 = wred_add(z * att[32 + lane]);
    z = xl[64 + lane] + xr[64 + lane];   z = z > 0.f ? z : 0.2f * z; lg2 = wred_add(z * att[64 + lane]);
    z = xl[96 + lane] + xr[96 + lane];   z = z > 0.f ? z : 0.2f * z; lg3 = wred_add(z * att[96 + lane]);
  }
  if (lane == 0) {
    float* m = MAXB + (size_t)d * NH;
    atomicMax(&m[0], lg0);
    atomicMax(&m[1], lg1);
    atomicMax(&m[2], lg2);
    atomicMax(&m[3], lg3);
  }
}

// ---------------------------------------------------- GATv2 edge pass 2: denom + numerator
__global__ void gat_acc_kernel(const int* __restrict__ src, const int* __restrict__ dst, int E,
                               const float* __restrict__ XL, const float* __restrict__ XR,
                               const float* __restrict__ att, const float* __restrict__ MAXB,
                               float* __restrict__ DEN, float* __restrict__ NUM) {
  int e = (int)(((size_t)blockIdx.x * blockDim.x + threadIdx.x) >> 5);
  int lane = threadIdx.x & 31;
  if (e >= E) return;
  int s = src[e], d = dst[e];
  const float* xl = XL + (size_t)s * HID;
  const float* xr = XR + (size_t)d * HID;
  float lg0, lg1, lg2, lg3;
  {
    float z;
    z = xl[lane] + xr[lane];             z = z > 0.f ? z : 0.2f * z; lg0 = wred_add(z * att[lane]);
    z = xl[32 + lane] + xr[32 + lane];   z = z > 0.f ? z : 0.2f * z; lg1 = wred_add(z * att[32 + lane]);
    z = xl[64 + lane] + xr[64 + lane];   z = z > 0.f ? z : 0.2f * z; lg2 = wred_add(z * att[64 + lane]);
    z = xl[96 + lane] + xr[96 + lane];   z = z > 0.f ? z : 0.2f * z; lg3 = wred_add(z * att[96 + lane]);
  }
  const float* m = MAXB + (size_t)d * NH;
  float e0 = __expf(lg0 - m[0]);
  float e1 = __expf(lg1 - m[1]);
  float e2 = __expf(lg2 - m[2]);
  float e3 = __expf(lg3 - m[3]);
  if (lane == 0) {
    float* dn = DEN + (size_t)d * NH;
    atomicAdd(&dn[0], e0);
    atomicAdd(&dn[1], e1);
    atomicAdd(&dn[2], e2);
    atomicAdd(&dn[3], e3);
  }
  float* num = NUM + (size_t)d * HID;
  atomicAdd(&num[lane],      e0 * xl[lane]);
  atomicAdd(&num[32 + lane], e1 * xl[32 + lane]);
  atomicAdd(&num[64 + lane], e2 * xl[64 + lane]);
  atomicAdd(&num[96 + lane], e3 * xl[96 + lane]);
}

// ---------------------------------------------------- Transformer edge pass 1 (max logit)
__global__ void trans_max_kernel(const int* __restrict__ src, const int* __restrict__ dst, int E,
                                 const float* __restrict__ Q, const float* __restrict__ K,
                                 const float* __restrict__ We, const float* __restrict__ attr,
                                 float* __restrict__ MAXB) {
  int e = (int)(((size_t)blockIdx.x * blockDim.x + threadIdx.x) >> 5);
  int lane = threadIdx.x & 31;
  if (e >= E) return;
  int s = src[e], d = dst[e];
  const float scale = 0.17677669529663687f;  // 1/sqrt(32)
  float a = attr[e];
  const float* q = Q + (size_t)d * HID;
  const float* k = K + (size_t)s * HID;
  float lg0 = wred_add(q[lane]      * (k[lane]      + a * We[lane]))      * scale;
  float lg1 = wred_add(q[32 + lane] * (k[32 + lane] + a * We[32 + lane])) * scale;
  float lg2 = wred_add(q[64 + lane] * (k[64 + lane] + a * We[64 + lane])) * scale;
  float lg3 = wred_add(q[96 + lane] * (k[96 + lane] + a * We[96 + lane])) * scale;
  if (lane == 0) {
    float* m = MAXB + (size_t)d * NH;
    atomicMax(&m[0], lg0);
    atomicMax(&m[1], lg1);
    atomicMax(&m[2], lg2);
    atomicMax(&m[3], lg3);
  }
}

// ---------------------------------------------------- Transformer edge pass 2
__global__ void trans_acc_kernel(const int* __restrict__ src, const int* __restrict__ dst, int E,
                                 const float* __restrict__ Q, const float* __restrict__ K,
                                 const float* __restrict__ V, const float* __restrict__ We,
                                 const float* __restrict__ attr, const float* __restrict__ MAXB,
                                 float* __restrict__ DEN, float* __restrict__ NUM) {
  int e = (int)(((size_t)blockIdx.x * blockDim.x + threadIdx.x) >> 5);
  int lane = threadIdx.x & 31;
  if (e >= E) return;
  int s = src[e], d = dst[e];
  const float scale = 0.17677669529663687f;
  float a = attr[e];
  const float* q = Q + (size_t)d * HID;
  const float* k = K + (size_t)s * HID;
  const float* v = V + (size_t)s * HID;
  float lg0 = wred_add(q[lane]      * (k[lane]      + a * We[lane]))      * scale;
  float lg1 = wred_add(q[32 + lane] * (k[32 + lane] + a * We[32 + lane])) * scale;
  float lg2 = wred_add(q[64 + lane] * (k[64 + lane] + a * We[64 + lane])) * scale;
  float lg3 = wred_add(q[96 + lane] * (k[96 + lane] + a * We[96 + lane])) * scale;
  const float* m = MAXB + (size_t)d * NH;
  float e0 = __expf(lg0 - m[0]);
  float e1 = __expf(lg1 - m[1]);
  float e2 = __expf(lg2 - m[2]);
  float e3 = __expf(lg3 - m[3]);
  if (lane == 0) {
    float* dn = DEN + (size_t)d * NH;
    atomicAdd(&dn[0], e0);
    atomicAdd(&dn[1], e1);
    atomicAdd(&dn[2], e2);
    atomicAdd(&dn[3], e3);
  }
  float* num = NUM + (size_t)d * HID;
  atomicAdd(&num[lane],      e0 * (v[lane]      + a * We[lane]));
  atomicAdd(&num[32 + lane], e1 * (v[32 + lane] + a * We[32 + lane]));
  atomicAdd(&num[64 + lane], e2 * (v[64 + lane] + a * We[64 + lane]));
  atomicAdd(&num[96 + lane], e3 * (v[96 + lane] + a * We[96 + lane]));
}

// ---------------------------------------------------- SAGE scatter: SUM[dst]+=X[src], CNT[dst]+=1
__global__ void sage_acc_kernel(const int* __restrict__ src, const int* __restrict__ dst, int E,
                                const float* __restrict__ X, float* __restrict__ SUM,
                                float* __restrict__ CNT) {
  int e = (int)(((size_t)blockIdx.x * blockDim.x + threadIdx.x) >> 5);
  int lane = threadIdx.x & 31;
  if (e >= E) return;
  int s = src[e], d = dst[e];
  const float* x = X + (size_t)s * HID;
  float* o = SUM + (size_t)d * HID;
  atomicAdd(&o[lane],      x[lane]);
  atomicAdd(&o[32 + lane], x[32 + lane]);
  atomicAdd(&o[64 + lane], x[64 + lane]);
  atomicAdd(&o[96 + lane], x[96 + lane]);
  if (lane == 0) atomicAdd(&CNT[d], 1.0f);
}

// ---------------------------------------------------- SUM -> mean (in place)
__global__ void mean_div_kernel(float* __restrict__ SUM, const float* __restrict__ CNT, int n) {
  size_t i = (size_t)blockIdx.x * blockDim.x + threadIdx.x;
  if (i >= (size_t)n * HID) return;
  float c = CNT[i >> 7];
  SUM[i] /= fmaxf(c, 1.0f);
}

// ---------------------------------------------------- FiLM gamma/beta (rank-1 cond)
__device__ __forceinline__ float gelu_exact(float x) {
  return 0.5f * x * (1.0f + erff(x * 0.70710678118654752f));
}
__global__ void film_params_kernel(const float* __restrict__ cond, const float* __restrict__ W1,
                                   const float* __restrict__ b1, const float* __restrict__ W2,
                                   const float* __restrict__ b2, float* __restrict__ gb) {
  __shared__ float h[HID];
  int t = threadIdx.x;  // 256 threads
  if (t < HID) {
    float a = b1[t];
#pragma unroll
    for (int i = 0; i < 5; ++i) a += cond[i] * W1[i * HID + t];
    h[t] = gelu_exact(a);
  }
  __syncthreads();
  float a = b2[t];
  for (int i = 0; i < HID; ++i) a += h[i] * W2[i * 256 + t];
  gb[t] = a;  // [0:128] gamma, [128:256] beta
}

// ---------------------------------------------------- final combine for ball nodes (+LN+FiLM)
__global__ void combine_ball_kernel(
    const float* __restrict__ NUMrel, const float* __restrict__ DENrel, const float* __restrict__ brel,
    const float* __restrict__ NUMrec, const float* __restrict__ DENrec, const float* __restrict__ SKIP,
    const float* __restrict__ DIS, const float* __restrict__ XRdis,
    const float* __restrict__ NUMcon, const float* __restrict__ DENcon, const float* __restrict__ bcon,
    const float* __restrict__ xin, const float* __restrict__ lng, const float* __restrict__ lnb,
    const float* __restrict__ gb, float* __restrict__ out, int N) {
  int i = (int)(((size_t)blockIdx.x * blockDim.x + threadIdx.x) >> 5);
  int lane = threadIdx.x & 31;
  if (i >= N) return;
  float v0, v1, v2, v3;
  const float* dr = DENrel + (size_t)i * NH;
  const float* dc = DENrec + (size_t)i * NH;
  const float* dq = DENcon + (size_t)i * NH;
#define BALL_VAL(j)                                                                     \
  (NUMrel[(size_t)i * HID + (j)*32 + lane] / (dr[j] + 1e-16f) + brel[(j)*32 + lane] +   \
   NUMrec[(size_t)i * HID + (j)*32 + lane] / (dc[j] + 1e-16f) +                         \
   SKIP[(size_t)i * HID + (j)*32 + lane] + DIS[(size_t)i * HID + (j)*32 + lane] +       \
   XRdis[(size_t)i * HID + (j)*32 + lane] +                                             \
   NUMcon[(size_t)i * HID + (j)*32 + lane] / (dq[j] + 1e-16f) + bcon[(j)*32 + lane] +   \
   xin[(size_t)i * HID + (j)*32 + lane])
  v0 = BALL_VAL(0);
  v1 = BALL_VAL(1);
  v2 = BALL_VAL(2);
  v3 = BALL_VAL(3);
#undef BALL_VAL
  float mu = wred_add(v0 + v1 + v2 + v3) * (1.0f / 128.0f);
  float q = (v0 - mu) * (v0 - mu) + (v1 - mu) * (v1 - mu) + (v2 - mu) * (v2 - mu) +
            (v3 - mu) * (v3 - mu);
  float var = wred_add(q) * (1.0f / 128.0f);
  float inv = rsqrtf(var + 1e-5f);
#pragma unroll
  for (int j = 0; j < 4; ++j) {
    float v = (j == 0) ? v0 : (j == 1) ? v1 : (j == 2) ? v2 : v3;
    int dim = j * 32 + lane;
    float nrm = (v - mu) * inv * lng[dim] + lnb[dim];
    out[(size_t)i * HID + dim] = gb[dim] * nrm + gb[HID + dim];
  }
}

// ---------------------------------------------------- final combine for context nodes
__global__ void combine_ctx_kernel(const float* __restrict__ INFO, const float* __restrict__ XRinf,
                                   const float* __restrict__ xin, const float* __restrict__ lng,
                                   const float* __restrict__ lnb, const float* __restrict__ gb,
                                   float* __restrict__ out, int N) {
  int i = (int)(((size_t)blockIdx.x * blockDim.x + threadIdx.x) >> 5);
  int lane = threadIdx.x & 31;
  if (i >= N) return;
  float v0 = INFO[(size_t)i * HID + lane]      + XRinf[(size_t)i * HID + lane]      + xin[(size_t)i * HID + lane];
  float v1 = INFO[(size_t)i * HID + 32 + lane] + XRinf[(size_t)i * HID + 32 + lane] + xin[(size_t)i * HID + 32 + lane];
  float v2 = INFO[(size_t)i * HID + 64 + lane] + XRinf[(size_t)i * HID + 64 + lane] + xin[(size_t)i * HID + 64 + lane];
  float v3 = INFO[(size_t)i * HID + 96 + lane] + XRinf[(size_t)i * HID + 96 + lane] + xin[(size_t)i * HID + 96 + lane];
  float mu = wred_add(v0 + v1 + v2 + v3) * (1.0f / 128.0f);
  float q = (v0 - mu) * (v0 - mu) + (v1 - mu) * (v1 - mu) + (v2 - mu) * (v2 - mu) +
            (v3 - mu) * (v3 - mu);
  float var = wred_add(q) * (1.0f / 128.0f);
  float inv = rsqrtf(var + 1e-5f);
#pragma unroll
  for (int j = 0; j < 4; ++j) {
    float v = (j == 0) ? v0 : (j == 1) ? v1 : (j == 2) ? v2 : v3;
    int dim = j * 32 + lane;
    float nrm = (v - mu) * inv * lng[dim] + lnb[dim];
    out[(size_t)i * HID + dim] = gb[dim] * nrm + gb[HID + dim];
  }
}

// ================================================================ host
extern "C" void kernel_launch(void* const* d_in, const int* in_sizes, int n_in,
                              void* d_out, int out_size, void* d_ws, size_t ws_size,
                              hipStream_t stream) {
  if (n_in < 48) return;
  // ---- inputs (insertion order of setup_inputs / nested params dicts) ----
  const float* x_ball = (const float*)d_in[0];
  const float* x_ctx  = (const float*)d_in[1];
  const float* cond   = (const float*)d_in[2];
  const float* eattr  = (const float*)d_in[3];
  // rel (gat): Wl bl Wr br att b -> 4..9
  const float *Wl_rel = (const float*)d_in[4],  *bl_rel = (const float*)d_in[5];
  const float *Wr_rel = (const float*)d_in[6],  *br_rel = (const float*)d_in[7];
  const float *att_rel = (const float*)d_in[8], *b_rel  = (const float*)d_in[9];
  // rec (trans): Wq bq Wk bk Wv bv We Wskip bskip -> 10..18
  const float *Wq = (const float*)d_in[10], *bq = (const float*)d_in[11];
  const float *Wk = (const float*)d_in[12], *bk = (const float*)d_in[13];
  const float *Wv = (const float*)d_in[14], *bv = (const float*)d_in[15];
  const float *We = (const float*)d_in[16];
  const float *Wskip = (const float*)d_in[17], *bskip = (const float*)d_in[18];
  // dis (sage): Wl bl Wr -> 19..21
  const float *Wl_dis = (const float*)d_in[19], *bl_dis = (const float*)d_in[20];
  const float *Wr_dis = (const float*)d_in[21];
  // con (gat): 22..27
  const float *Wl_con = (const float*)d_in[22], *bl_con = (const float*)d_in[23];
  const float *Wr_con = (const float*)d_in[24], *br_con = (const float*)d_in[25];
  const float *att_con = (const float*)d_in[26], *b_con = (const float*)d_in[27];
  // inf (sage): 28..30
  const float *Wl_inf = (const float*)d_in[28], *bl_inf = (const float*)d_in[29];
  const float *Wr_inf = (const float*)d_in[30];
  // ln: 31..34
  const float *lng_ball = (const float*)d_in[31], *lnb_ball = (const float*)d_in[32];
  const float *lng_ctx  = (const float*)d_in[33], *lnb_ctx  = (const float*)d_in[34];
  // film: 35..42
  const float *fW1b = (const float*)d_in[35], *fb1b = (const float*)d_in[36];
  const float *fW2b = (const float*)d_in[37], *fb2b = (const float*)d_in[38];
  const float *fW1c = (const float*)d_in[39], *fb1c = (const float*)d_in[40];
  const float *fW2c = (const float*)d_in[41], *fb2c = (const float*)d_in[42];
  // edges: [2,E] each, src row then dst row
  const int NB = in_sizes[0] / HID;
  const int NC = in_sizes[1] / HID;
  const int Erel = in_sizes[43] / 2, Erec = in_sizes[44] / 2, Edis = in_sizes[45] / 2,
            Einf = in_sizes[46] / 2, Econ = in_sizes[47] / 2;
  const int *s_rel = (const int*)d_in[43], *d_rel = s_rel + Erel;
  const int *s_rec = (const int*)d_in[44], *d_rec = s_rec + Erec;
  const int *s_dis = (const int*)d_in[45], *d_dis = s_dis + Edis;
  const int *s_inf = (const int*)d_in[46], *d_inf = s_inf + Einf;
  const int *s_con = (const int*)d_in[47], *d_con = s_con + Econ;

  // ---- workspace carve-out (floats) ----
  float* ws = (float*)d_ws;
  size_t off = 0;
  auto alloc = [&](size_t n) { float* p = ws + off; off += n; return p; };
  const size_t NBF = (size_t)NB * HID, NCF = (size_t)NC * HID;
  float* XLrel = alloc(NBF);
  float* XRrel = alloc(NBF);
  float* Qb    = alloc(NBF);
  float* Kb    = alloc(NBF);
  float* Vb    = alloc(NBF);
  float* SKIPb = alloc(NBF);
  float* XRdis = alloc(NBF);
  float* XRcon = alloc(NBF);
  float* NUMrel = alloc(NBF);
  float* NUMrec = alloc(NBF);
  float* NUMcon = alloc(NBF);
  float* SUMdis = alloc(NBF);      // sum -> mean -> (in-place GEMM) dis output
  float* XLcon = alloc(NCF);
  float* XRinf = alloc(NCF);
  float* SUMinf = alloc(NCF);      // sum -> mean -> (in-place GEMM) inf output
  float* MAXrel = alloc((size_t)NB * NH);
  float* DENrel = alloc((size_t)NB * NH);
  float* MAXrec = alloc((size_t)NB * NH);
  float* DENrec = alloc((size_t)NB * NH);
  float* MAXcon = alloc((size_t)NB * NH);
  float* DENcon = alloc((size_t)NB * NH);
  float* CNTdis = alloc((size_t)NB);
  float* CNTinf = alloc((size_t)NC);
  float* GBball = alloc(256);
  float* GBctx  = alloc(256);
  (void)ws_size;

  const int TB = 256;
  auto gridN = [](size_t n, size_t per) { return (unsigned)((n + per - 1) / per); };

  // ---- FiLM parameters (rank-1 cond) ----
  film_params_kernel<<<1, 256, 0, stream>>>(cond, fW1b, fb1b, fW2b, fb2b, GBball);
  film_params_kernel<<<1, 256, 0, stream>>>(cond, fW1c, fb1c, fW2c, fb2c, GBctx);

  // ---- clear accumulators, init maxes to -inf ----
  fill_kernel<<<2048, TB, 0, stream>>>(NUMrel, NBF * 4 + NCF, 0.0f);  // NUMrel..SUMdis + SUMinf-range handled below
  fill_kernel<<<2048, TB, 0, stream>>>(SUMinf, NCF, 0.0f);
  fill_kernel<<<512, TB, 0, stream>>>(DENrel, (size_t)NB * NH, 0.0f);
  fill_kernel<<<512, TB, 0, stream>>>(DENrec, (size_t)NB * NH, 0.0f);
  fill_kernel<<<512, TB, 0, stream>>>(DENcon, (size_t)NB * NH, 0.0f);
  fill_kernel<<<256, TB, 0, stream>>>(CNTdis, (size_t)NB, 0.0f);
  fill_kernel<<<64, TB, 0, stream>>>(CNTinf, (size_t)NC, 0.0f);
  fill_kernel<<<512, TB, 0, stream>>>(MAXrel, (size_t)NB * NH, -INFINITY);
  fill_kernel<<<512, TB, 0, stream>>>(MAXrec, (size_t)NB * NH, -INFINITY);
  fill_kernel<<<512, TB, 0, stream>>>(MAXcon, (size_t)NB * NH, -INFINITY);

  // ---- node projections via f32 WMMA GEMMs ----
  unsigned gB = gridN(NB, 128), gC = gridN(NC, 128);
  gemm128_wmma<<<gB, TB, 0, stream>>>(x_ball, Wl_rel, bl_rel, XLrel, NB);
  gemm128_wmma<<<gB, TB, 0, stream>>>(x_ball, Wr_rel, br_rel, XRrel, NB);
  gemm128_wmma<<<gB, TB, 0, stream>>>(x_ball, Wq, bq, Qb, NB);
  gemm128_wmma<<<gB, TB, 0, stream>>>(x_ball, Wk, bk, Kb, NB);
  gemm128_wmma<<<gB, TB, 0, stream>>>(x_ball, Wv, bv, Vb, NB);
  gemm128_wmma<<<gB, TB, 0, stream>>>(x_ball, Wskip, bskip, SKIPb, NB);
  gemm128_wmma<<<gB, TB, 0, stream>>>(x_ball, Wr_dis, nullptr, XRdis, NB);
  gemm128_wmma<<<gB, TB, 0, stream>>>(x_ball, Wr_con, br_con, XRcon, NB);
  gemm128_wmma<<<gC, TB, 0, stream>>>(x_ctx, Wl_con, bl_con, XLcon, NC);
  gemm128_wmma<<<gC, TB, 0, stream>>>(x_ctx, Wr_inf, nullptr, XRinf, NC);

  // ---- attention branches: two edge passes each (wave per edge) ----
  gat_max_kernel<<<gridN(Erel, 8), TB, 0, stream>>>(s_rel, d_rel, Erel, XLrel, XRrel, att_rel, MAXrel);
  gat_acc_kernel<<<gridN(Erel, 8), TB, 0, stream>>>(s_rel, d_rel, Erel, XLrel, XRrel, att_rel, MAXrel, DENrel, NUMrel);
  trans_max_kernel<<<gridN(Erec, 8), TB, 0, stream>>>(s_rec, d_rec, Erec, Qb, Kb, We, eattr, MAXrec);
  trans_acc_kernel<<<gridN(Erec, 8), TB, 0, stream>>>(s_rec, d_rec, Erec, Qb, Kb, Vb, We, eattr, MAXrec, DENrec, NUMrec);
  gat_max_kernel<<<gridN(Econ, 8), TB, 0, stream>>>(s_con, d_con, Econ, XLcon, XRcon, att_con, MAXcon);
  gat_acc_kernel<<<gridN(Econ, 8), TB, 0, stream>>>(s_con, d_con, Econ, XLcon, XRcon, att_con, MAXcon, DENcon, NUMcon);

  // ---- SAGE branches: scatter-sum, mean, then mean @ Wl + bl (in-place WMMA GEMM) ----
  sage_acc_kernel<<<gridN(Edis, 8), TB, 0, stream>>>(s_dis, d_dis, Edis, x_ball, SUMdis, CNTdis);
  sage_acc_kernel<<<gridN(Einf, 8), TB, 0, stream>>>(s_inf, d_inf, Einf, x_ball, SUMinf, CNTinf);
  mean_div_kernel<<<gridN(NBF, TB), TB, 0, stream>>>(SUMdis, CNTdis, NB);
  mean_div_kernel<<<gridN(NCF, TB), TB, 0, stream>>>(SUMinf, CNTinf, NC);
  gemm128_wmma<<<gB, TB, 0, stream>>>(SUMdis, Wl_dis, bl_dis, SUMdis, NB);  // safe: strip-local in-place
  gemm128_wmma<<<gC, TB, 0, stream>>>(SUMinf, Wl_inf, bl_inf, SUMinf, NC);

  // ---- combine + residual + LayerNorm + FiLM ----
  float* out_ball = (float*)d_out;
  float* out_ctx  = (float*)d_out + NBF;
  combine_ball_kernel<<<gridN(NB, 8), TB, 0, stream>>>(
      NUMrel, DENrel, b_rel, NUMrec, DENrec, SKIPb, SUMdis, XRdis, NUMcon, DENcon, b_con,
      x_ball, lng_ball, lnb_ball, GBball, out_ball, NB);
  combine_ctx_kernel<<<gridN(NC, 8), TB, 0, stream>>>(
      SUMinf, XRinf, x_ctx, lng_ctx, lnb_ctx, GBctx, out_ctx, NC);
}